// Combine_18622978196105
// MI455X (gfx1250) — compile-verified
//
#include <hip/hip_runtime.h>

#define SLOPE 0.2f
#define BN_EPS 1e-5f

typedef __attribute__((ext_vector_type(2))) float v2f;
typedef __attribute__((ext_vector_type(8))) float v8f;

static __device__ __forceinline__ v8f wmma4(v2f a, v2f b, v8f c) {
  // D = A(16x4 f32) * B(4x16 f32) + C(16x16 f32)
  return __builtin_amdgcn_wmma_f32_16x16x4_f32(false, a, false, b, (short)0, c,
                                               false, false);
}

// ======================= WMMA implicit-GEMM Conv2d ==========================
// k=4 s=2 p=1, no bias.  GEMM view per batch: M=Cout, N=Hout*Wout, K=Cin*16.
// One wave computes MT 16x16 output tiles stacked in M (sharing B gathers).
// Gathers are branch-free: clamped address + 0/1 float mask (no EXEC churn).
template <int MT>
__global__ void __launch_bounds__(32) k_conv_down_wmma(
    const float* __restrict__ in, const float* __restrict__ w,
    float* __restrict__ out, int B, int Cin, int Cout, int Hin, int Win,
    int act) {
  int Hout = Hin >> 1, Wout = Win >> 1;
  int ntiles = (Hout * Wout) >> 4;
  int cgroups = (Cout >> 4) / MT;
  int t = blockIdx.x;
  int nt = t % ntiles;
  t /= ntiles;
  int cg = t % cgroups;
  int b = t / cgroups;
  int co0 = cg * MT * 16;
  int lane = threadIdx.x;
  int q = lane & 15, half = lane >> 4;
  int p = (nt << 4) + q;  // this lane's output pixel (B col); A row m=q
  int y = p / Wout, x = p - y * Wout;

  // this lane-half owns kx = 2*half and 2*half+1
  int ixa = 2 * x + 2 * half - 1;
  int ixb = ixa + 1;
  bool vxa = (unsigned)ixa < (unsigned)Win;
  bool vxb = (unsigned)ixb < (unsigned)Win;

  int offa[4], offb[4];
  float ma[4], mb[4];
#pragma unroll
  for (int ky = 0; ky < 4; ++ky) {
    int iy = 2 * y + ky - 1;
    bool vy = (unsigned)iy < (unsigned)Hin;
    int base = vy ? iy * Win : 0;
    bool va = vy && vxa;
    bool vb = vy && vxb;
    offa[ky] = va ? base + ixa : 0;
    offb[ky] = vb ? base + ixb : 0;
    ma[ky] = va ? 1.f : 0.f;
    mb[ky] = vb ? 1.f : 0.f;
  }

  const float* wrow[MT];
#pragma unroll
  for (int m = 0; m < MT; ++m)
    wrow[m] = w + (size_t)(co0 + m * 16 + q) * Cin * 16;

  v8f acc[MT];
#pragma unroll
  for (int m = 0; m < MT; ++m) acc[m] = (v8f){};

  for (int ci = 0; ci < Cin; ++ci) {
    const float* ip = in + (size_t)(b * Cin + ci) * Hin * Win;
#pragma unroll
    for (int ky = 0; ky < 4; ++ky) {  // kk = ci*16 + ky*4 + (2*half + v)
      v2f bb;
      bb.x = ip[offa[ky]] * ma[ky];
      bb.y = ip[offb[ky]] * mb[ky];
#pragma unroll
      for (int m = 0; m < MT; ++m) {
        v2f a = *(const v2f*)(wrow[m] + ci * 16 + 4 * ky + 2 * half);
        acc[m] = wmma4(a, bb, acc[m]);
      }
    }
  }
#pragma unroll
  for (int m = 0; m < MT; ++m) {
#pragma unroll
    for (int v = 0; v < 8; ++v) {  // D: M = v + 8*half, N = q
      float val = acc[m][v];
      if (act == 1) val = val > 0.f ? val : SLOPE * val;
      int co = co0 + m * 16 + v + 8 * half;
      out[((size_t)(b * Cout + co) * Hout + y) * Wout + x] = val;
    }
  }
}

// ================== WMMA implicit-GEMM ConvTranspose2d ======================
// k=4 s=2 p=1, torch weight layout [Cin,Cout,4,4].  Tap (ky,kx) valid iff
// (y+1-ky) and (x+1-kx) are even and in range; predicates hoisted, branch-free.
// Writes channels [0,Cout) of a concat buffer with channel capacity ccap.
template <int MT>
__global__ void __launch_bounds__(32) k_conv_up_wmma(
    const float* __restrict__ in, const float* __restrict__ w,
    float* __restrict__ out, int B, int Cin, int Cout, int Hin, int Win,
    int ccap) {
  int Hout = Hin << 1, Wout = Win << 1;
  int ntiles = (Hout * Wout) >> 4;
  int cgroups = (Cout >> 4) / MT;
  int t = blockIdx.x;
  int nt = t % ntiles;
  t /= ntiles;
  int cg = t % cgroups;
  int b = t / cgroups;
  int co0 = cg * MT * 16;
  int lane = threadIdx.x;
  int q = lane & 15, half = lane >> 4;
  int p = (nt << 4) + q;
  int y = p / Wout, x = p - y * Wout;

  int txa = x + 1 - 2 * half;  // kx = 2*half
  int ixa = txa >> 1;
  bool vxa = ((txa & 1) == 0) && ((unsigned)ixa < (unsigned)Win);
  int txb = txa - 1;  // kx = 2*half + 1
  int ixb = txb >> 1;
  bool vxb = ((txb & 1) == 0) && ((unsigned)ixb < (unsigned)Win);

  int offa[4], offb[4];
  float ma[4], mb[4];
#pragma unroll
  for (int ky = 0; ky < 4; ++ky) {
    int ty = y + 1 - ky;
    int iy = ty >> 1;
    bool vy = ((ty & 1) == 0) && ((unsigned)iy < (unsigned)Hin);
    int base = vy ? iy * Win : 0;
    bool va = vy && vxa;
    bool vb = vy && vxb;
    offa[ky] = va ? base + ixa : 0;
    offb[ky] = vb ? base + ixb : 0;
    ma[ky] = va ? 1.f : 0.f;
    mb[ky] = vb ? 1.f : 0.f;
  }

  size_t wlane[MT];
#pragma unroll
  for (int m = 0; m < MT; ++m) wlane[m] = (size_t)(co0 + m * 16 + q) * 16;

  v8f acc[MT];
#pragma unroll
  for (int m = 0; m < MT; ++m) acc[m] = (v8f){};

  for (int ci = 0; ci < Cin; ++ci) {
    const float* ip = in + (size_t)(b * Cin + ci) * Hin * Win;
    const float* wci = w + (size_t)ci * Cout * 16;
#pragma unroll
    for (int ky = 0; ky < 4; ++ky) {
      v2f bb;
      bb.x = ip[offa[ky]] * ma[ky];
      bb.y = ip[offb[ky]] * mb[ky];
#pragma unroll
      for (int m = 0; m < MT; ++m) {
        v2f a = *(const v2f*)(wci + wlane[m] + 4 * ky + 2 * half);
        acc[m] = wmma4(a, bb, acc[m]);
      }
    }
  }
#pragma unroll
  for (int m = 0; m < MT; ++m) {
#pragma unroll
    for (int v = 0; v < 8; ++v) {
      int co = co0 + m * 16 + v + 8 * half;
      out[((size_t)(b * ccap + co) * Hout + y) * Wout + x] = acc[m][v];
    }
  }
}

// -------- per-channel mean/biased-var over (N,H,W); one block per channel ---
__global__ void k_chan_stats(const float* __restrict__ xb,
                             float* __restrict__ mean, float* __restrict__ var,
                             int B, int H, int W, int ccap) {
  int c = blockIdx.x;
  int HW = H * W;
  int n = B * HW;
  float s = 0.f, ss = 0.f;
  for (int i = threadIdx.x; i < n; i += blockDim.x) {
    int b = i / HW;
    int p = i - b * HW;
    float v = xb[(size_t)(b * ccap + c) * HW + p];
    s += v;
    ss += v * v;
  }
  __shared__ float s1[256], s2[256];
  s1[threadIdx.x] = s;
  s2[threadIdx.x] = ss;
  __syncthreads();
  for (int st = 128; st > 0; st >>= 1) {
    if (threadIdx.x < st) {
      s1[threadIdx.x] += s1[threadIdx.x + st];
      s2[threadIdx.x] += s2[threadIdx.x + st];
    }
    __syncthreads();
  }
  if (threadIdx.x == 0) {
    float m = s1[0] / (float)n;
    mean[c] = m;
    var[c] = s2[0] / (float)n - m * m;
  }
}

// ---------------- in-place BN + activation (1=lrelu, 2=relu) ----------------
__global__ void k_bn_act(float* __restrict__ xb, const float* __restrict__ mean,
                         const float* __restrict__ var,
                         const float* __restrict__ g,
                         const float* __restrict__ bt, int B, int C, int H,
                         int W, int ccap, int act) {
  int total = B * C * H * W;
  int idx = blockIdx.x * blockDim.x + threadIdx.x;
  if (idx >= total) return;
  int HW = H * W;
  int p = idx % HW;
  int t = idx / HW;
  int c = t % C;
  int b = t / C;
  size_t off = (size_t)(b * ccap + c) * HW + p;
  float inv = rsqrtf(var[c] + BN_EPS);
  float v = (xb[off] - mean[c]) * inv * g[c] + bt[c];
  if (act == 1)
    v = v > 0.f ? v : SLOPE * v;
  else
    v = v > 0.f ? v : 0.f;
  xb[off] = v;
}

// --------- copy skip tensor [B,C,H,W] into concat channels [coff,coff+C) ----
__global__ void k_copy_skip(const float* __restrict__ skip,
                            float* __restrict__ dst, int B, int C, int H, int W,
                            int ccap, int coff) {
  int total = B * C * H * W;
  int idx = blockIdx.x * blockDim.x + threadIdx.x;
  if (idx >= total) return;
  int HW = H * W;
  int p = idx % HW;
  int t = idx / HW;
  int c = t % C;
  int b = t / C;
  dst[(size_t)(b * ccap + coff + c) * HW + p] = skip[idx];
}

// --- nearest x2 upsample + ZeroPad2d((1,0,1,0)) + conv k4 p1 + bias + tanh --
// u45: [8,64,128,128] -> u_out: [8,3,256,256] (written to d_out). Branch-free.
__global__ void k_final(const float* __restrict__ u45,
                        const float* __restrict__ wfin,
                        const float* __restrict__ bfin,
                        float* __restrict__ out) {
  int total = 8 * 3 * 256 * 256;
  int idx = blockIdx.x * blockDim.x + threadIdx.x;
  if (idx >= total) return;
  int x = idx & 255;
  int t = idx >> 8;
  int y = t & 255;
  t >>= 8;
  int c = t % 3;
  int b = t / 3;
  int uyo[4], uxo[4];
  float my[4], mx[4];
#pragma unroll
  for (int k = 0; k < 4; ++k) {
    int py = y + k - 1;  // index into 257-wide padded plane; py==0 = zero pad
    bool vy = (py >= 1) && (py <= 256);
    uyo[k] = vy ? ((py - 1) >> 1) * 128 : 0;
    my[k] = vy ? 1.f : 0.f;
    int px = x + k - 1;
    bool vx = (px >= 1) && (px <= 256);
    uxo[k] = vx ? ((px - 1) >> 1) : 0;
    mx[k] = vx ? 1.f : 0.f;
  }
  float acc = 0.f;
  for (int ci = 0; ci < 64; ++ci) {
    const float* up = u45 + (size_t)(b * 64 + ci) * 128 * 128;
    const float* wc = wfin + (size_t)(c * 64 + ci) * 16;
#pragma unroll
    for (int ky = 0; ky < 4; ++ky) {
#pragma unroll
      for (int kx = 0; kx < 4; ++kx) {
        acc = fmaf(up[uyo[ky] + uxo[kx]], wc[ky * 4 + kx] * my[ky] * mx[kx],
                   acc);
      }
    }
  }
  out[idx] = tanhf(acc + bfin[c]);
}

// ---------------- DFT stage 1: R = U @ W^T for Wr and Wi (WMMA f32) ---------
__global__ void __launch_bounds__(32) k_dft_rows(const float* __restrict__ U,
                                                 const float* __restrict__ Wr,
                                                 const float* __restrict__ Wi,
                                                 float* __restrict__ Rr,
                                                 float* __restrict__ Ri) {
  int tile = blockIdx.x;  // 384 * 16 tiles
  int nt = tile & 15;
  int mt = tile >> 4;
  int r0 = mt << 4, n0 = nt << 4;
  int lane = threadIdx.x;
  int q = lane & 15;
  int half = lane >> 4;
  v8f rr = {};
  v8f ri = {};
  const float* arow = U + (size_t)(r0 + q) * 256;
  const float* brrow = Wr + (size_t)(n0 + q) * 256;  // B[k][n] = Wr[n][k]
  const float* birow = Wi + (size_t)(n0 + q) * 256;
  for (int k = 0; k < 256; k += 4) {
    int kb = k + 2 * half;
    v2f a = *(const v2f*)(arow + kb);
    v2f br = *(const v2f*)(brrow + kb);
    v2f bi = *(const v2f*)(birow + kb);
    rr = wmma4(a, br, rr);
    ri = wmma4(a, bi, ri);
  }
#pragma unroll
  for (int v = 0; v < 8; ++v) {
    int row = r0 + v + 8 * half;
    int col = n0 + q;
    Rr[(size_t)row * 256 + col] = rr[v];
    Ri[(size_t)row * 256 + col] = ri[v];
  }
}

// --- DFT stage 2: per image, Cr = Wr^T@Rr - Wi^T@Ri ; Ci = Wr^T@Ri + Wi^T@Rr
__global__ void __launch_bounds__(32) k_dft_cols(const float* __restrict__ Rr,
                                                 const float* __restrict__ Ri,
                                                 const float* __restrict__ Wr,
                                                 const float* __restrict__ Wi,
                                                 float* __restrict__ res) {
  int tile = blockIdx.x;  // 24 images * 256 tiles
  int img = tile >> 8;
  int t = tile & 255;
  int mt = t >> 4, nt = t & 15;
  int m0 = mt << 4, n0 = nt << 4;
  int lane = threadIdx.x;
  int q = lane & 15;
  int half = lane >> 4;
  const float* rrb = Rr + (size_t)img * 65536;
  const float* rib = Ri + (size_t)img * 65536;
  v8f cr = {};
  v8f ci = {};
  for (int k = 0; k < 256; k += 4) {
    int kb = k + 2 * half;
    v2f awr = {Wr[(size_t)kb * 256 + m0 + q],
               Wr[(size_t)(kb + 1) * 256 + m0 + q]};
    v2f awi = {Wi[(size_t)kb * 256 + m0 + q],
               Wi[(size_t)(kb + 1) * 256 + m0 + q]};
    v2f awin = {-awi.x, -awi.y};
    v2f brr = {rrb[(size_t)kb * 256 + n0 + q],
               rrb[(size_t)(kb + 1) * 256 + n0 + q]};
    v2f bri = {rib[(size_t)kb * 256 + n0 + q],
               rib[(size_t)(kb + 1) * 256 + n0 + q]};
    cr = wmma4(awr, brr, cr);
    cr = wmma4(awin, bri, cr);
    ci = wmma4(awr, bri, ci);
    ci = wmma4(awi, brr, ci);
  }
#pragma unroll
  for (int v = 0; v < 8; ++v) {
    int row = m0 + v + 8 * half;
    size_t base = (size_t)img * 256 * 512 + (size_t)row * 512;
    res[base + n0 + q] = cr[v];
    res[base + 256 + n0 + q] = ci[v];
  }
}

// ============================================================================
extern "C" void kernel_launch(void* const* d_in, const int* in_sizes, int n_in,
                              void* d_out, int out_size, void* d_ws,
                              size_t ws_size, hipStream_t stream) {
  (void)in_sizes;
  (void)out_size;
  (void)ws_size;
  bool flat = (n_in >= 28);
  int i = 0;
  const float* x = (const float*)d_in[i++];
  const float* w_d1 = (const float*)d_in[i++];
  const float* w_d2 = (const float*)d_in[i++];
  const float *g_d2, *b_d2;
  if (flat) { g_d2 = (const float*)d_in[i++]; b_d2 = (const float*)d_in[i++]; }
  else { const float* t = (const float*)d_in[i++]; g_d2 = t; b_d2 = t + 128; }
  const float* w_d3 = (const float*)d_in[i++];
  const float *g_d3, *b_d3;
  if (flat) { g_d3 = (const float*)d_in[i++]; b_d3 = (const float*)d_in[i++]; }
  else { const float* t = (const float*)d_in[i++]; g_d3 = t; b_d3 = t + 256; }
  const float* w_d4 = (const float*)d_in[i++];
  const float *g_d4, *b_d4;
  if (flat) { g_d4 = (const float*)d_in[i++]; b_d4 = (const float*)d_in[i++]; }
  else { const float* t = (const float*)d_in[i++]; g_d4 = t; b_d4 = t + 256; }
  const float* w_d5 = (const float*)d_in[i++];
  const float* w_u1 = (const float*)d_in[i++];
  const float *g_u1, *b_u1;
  if (flat) { g_u1 = (const float*)d_in[i++]; b_u1 = (const float*)d_in[i++]; }
  else { const float* t = (const float*)d_in[i++]; g_u1 = t; b_u1 = t + 256; }
  const float* w_u2 = (const float*)d_in[i++];
  const float *g_u2, *b_u2;
  if (flat) { g_u2 = (const float*)d_in[i++]; b_u2 = (const float*)d_in[i++]; }
  else { const float* t = (const float*)d_in[i++]; g_u2 = t; b_u2 = t + 256; }
  const float* w_u3 = (const float*)d_in[i++];
  const float *g_u3, *b_u3;
  if (flat) { g_u3 = (const float*)d_in[i++]; b_u3 = (const float*)d_in[i++]; }
  else { const float* t = (const float*)d_in[i++]; g_u3 = t; b_u3 = t + 128; }
  const float* w_u4 = (const float*)d_in[i++];
  const float *g_u4, *b_u4;
  if (flat) { g_u4 = (const float*)d_in[i++]; b_u4 = (const float*)d_in[i++]; }
  else { const float* t = (const float*)d_in[i++]; g_u4 = t; b_u4 = t + 32; }
  const float* w_fin = (const float*)d_in[i++];
  const float* b_fin = (const float*)d_in[i++];
  const float* Wr = (const float*)d_in[i++];
  const float* Wi = (const float*)d_in[i++];

  // ---- workspace layout (floats) ----
  float* ws = (float*)d_ws;
  size_t o = 0;
  float* d1 = ws + o;  o += (size_t)8 * 32 * 128 * 128;
  float* d2 = ws + o;  o += (size_t)8 * 128 * 64 * 64;
  float* d3 = ws + o;  o += (size_t)8 * 256 * 32 * 32;
  float* d4 = ws + o;  o += (size_t)8 * 256 * 16 * 16;
  float* d5 = ws + o;  o += (size_t)8 * 256 * 8 * 8;
  float* u1c = ws + o; o += (size_t)8 * 512 * 16 * 16;
  float* u2c = ws + o; o += (size_t)8 * 512 * 32 * 32;
  float* u3c = ws + o; o += (size_t)8 * 256 * 64 * 64;
  float* u4c = ws + o; o += (size_t)8 * 64 * 128 * 128;
  float* mean = ws + o; o += 256;
  float* var = ws + o;  o += 256;
  float* Rr = d2;  // dead after u3-stage skip copy; reuse
  float* Ri = d3;  // dead after u2-stage skip copy; reuse

  float* uout = (float*)d_out;                             // [8,3,256,256]
  float* res = (float*)d_out + (size_t)8 * 3 * 256 * 256;  // [8,3,256,512]

  const int TB = 256;
  auto G = [](int total) { return (total + 255) / 256; };
  // conv wmma grids: B * (Cout/16/MT) * (Hout*Wout/16) single-wave blocks
  auto CG = [](int B, int Cout, int Hout, int Wout) {
    return B * ((Cout >> 4) / 2) * ((Hout * Wout) >> 4);
  };

  // ================= encoder =================
  k_conv_down_wmma<2><<<CG(8, 32, 128, 128), 32, 0, stream>>>(
      x, w_d1, d1, 8, 3, 32, 256, 256, 1);
  k_conv_down_wmma<2><<<CG(8, 128, 64, 64), 32, 0, stream>>>(
      d1, w_d2, d2, 8, 32, 128, 128, 128, 0);
  k_chan_stats<<<128, TB, 0, stream>>>(d2, mean, var, 8, 64, 64, 128);
  k_bn_act<<<G(8 * 128 * 64 * 64), TB, 0, stream>>>(d2, mean, var, g_d2, b_d2,
                                                    8, 128, 64, 64, 128, 1);
  k_conv_down_wmma<2><<<CG(8, 256, 32, 32), 32, 0, stream>>>(
      d2, w_d3, d3, 8, 128, 256, 64, 64, 0);
  k_chan_stats<<<256, TB, 0, stream>>>(d3, mean, var, 8, 32, 32, 256);
  k_bn_act<<<G(8 * 256 * 32 * 32), TB, 0, stream>>>(d3, mean, var, g_d3, b_d3,
                                                    8, 256, 32, 32, 256, 1);
  k_conv_down_wmma<2><<<CG(8, 256, 16, 16), 32, 0, stream>>>(
      d3, w_d4, d4, 8, 256, 256, 32, 32, 0);
  k_chan_stats<<<256, TB, 0, stream>>>(d4, mean, var, 8, 16, 16, 256);
  k_bn_act<<<G(8 * 256 * 16 * 16), TB, 0, stream>>>(d4, mean, var, g_d4, b_d4,
                                                    8, 256, 16, 16, 256, 1);
  k_conv_down_wmma<2><<<CG(8, 256, 8, 8), 32, 0, stream>>>(
      d4, w_d5, d5, 8, 256, 256, 16, 16, 1);

  // ================= decoder =================
  k_conv_up_wmma<2><<<CG(8, 256, 16, 16), 32, 0, stream>>>(
      d5, w_u1, u1c, 8, 256, 256, 8, 8, 512);
  k_chan_stats<<<256, TB, 0, stream>>>(u1c, mean, var, 8, 16, 16, 512);
  k_bn_act<<<G(8 * 256 * 16 * 16), TB, 0, stream>>>(u1c, mean, var, g_u1, b_u1,
                                                    8, 256, 16, 16, 512, 2);
  k_copy_skip<<<G(8 * 256 * 16 * 16), TB, 0, stream>>>(d4, u1c, 8, 256, 16, 16,
                                                       512, 256);
  k_conv_up_wmma<2><<<CG(8, 256, 32, 32), 32, 0, stream>>>(
      u1c, w_u2, u2c, 8, 512, 256, 16, 16, 512);
  k_chan_stats<<<256, TB, 0, stream>>>(u2c, mean, var, 8, 32, 32, 512);
  k_bn_act<<<G(8 * 256 * 32 * 32), TB, 0, stream>>>(u2c, mean, var, g_u2, b_u2,
                                                    8, 256, 32, 32, 512, 2);
  k_copy_skip<<<G(8 * 256 * 32 * 32), TB, 0, stream>>>(d3, u2c, 8, 256, 32, 32,
                                                       512, 256);
  k_conv_up_wmma<2><<<CG(8, 128, 64, 64), 32, 0, stream>>>(
      u2c, w_u3, u3c, 8, 512, 128, 32, 32, 256);
  k_chan_stats<<<128, TB, 0, stream>>>(u3c, mean, var, 8, 64, 64, 256);
  k_bn_act<<<G(8 * 128 * 64 * 64), TB, 0, stream>>>(u3c, mean, var, g_u3, b_u3,
                                                    8, 128, 64, 64, 256, 2);
  k_copy_skip<<<G(8 * 128 * 64 * 64), TB, 0, stream>>>(d2, u3c, 8, 128, 64, 64,
                                                       256, 128);
  k_conv_up_wmma<2><<<CG(8, 32, 128, 128), 32, 0, stream>>>(
      u3c, w_u4, u4c, 8, 256, 32, 64, 64, 64);
  k_chan_stats<<<32, TB, 0, stream>>>(u4c, mean, var, 8, 128, 128, 64);
  k_bn_act<<<G(8 * 32 * 128 * 128), TB, 0, stream>>>(u4c, mean, var, g_u4, b_u4,
                                                     8, 32, 128, 128, 64, 2);
  k_copy_skip<<<G(8 * 32 * 128 * 128), TB, 0, stream>>>(d1, u4c, 8, 32, 128,
                                                        128, 64, 32);

  // ================= final conv + tanh -> u_out =================
  k_final<<<G(8 * 3 * 256 * 256), TB, 0, stream>>>(u4c, w_fin, b_fin, uout);

  // ================= DFT via WMMA f32 16x16x4 =================
  k_dft_rows<<<6144, 32, 0, stream>>>(uout, Wr, Wi, Rr, Ri);
  k_dft_cols<<<6144, 32, 0, stream>>>(Rr, Ri, Wr, Wi, res);
}